// R3GANInterpolativeDownsampler_86371792322784
// MI455X (gfx1250) — compile-verified
//
#include <hip/hip_runtime.h>
#include <cstdint>

// Depthwise 4x4 binomial blur + stride-2 downsample.
// B=8, C=128, H=256, W=256 -> OH=OW=128.  Memory-bound (AI ~1.6 FLOP/B),
// so we use the CDNA5 async global->LDS DMA path for data movement and
// plain VALU FMAs for the 16-tap stencil.

#define TILE_OH    8                 // output rows per workgroup
#define TILE_ROWS  (2*TILE_OH + 2)   // 18 input rows needed (stride 2, 4-tap)
#define ROW_STRIDE 264               // floats; >= 261, keeps b128 dests 16B aligned
#define PAD_L      4                 // LDS col = iw + PAD_L (iw = -1 lands at col 3)
#define NTHREADS   256

__global__ __launch_bounds__(NTHREADS) void blurpool_ds2_kernel(
    const float* __restrict__ x,     // (B*C, H, W)
    const float* __restrict__ kw,    // (4,4) kernel
    float* __restrict__ y,           // (B*C, OH, OW)
    int H, int W, int OH, int OW)
{
    __shared__ float tile[TILE_ROWS * ROW_STRIDE];   // 19,008 B

    const int tid   = threadIdx.x;
    const int plane = blockIdx.y;              // b*C + c
    const int oh0   = blockIdx.x * TILE_OH;
    const int ih0   = 2 * oh0 - 1;             // padding radius = 1

    const float* xp = x + (size_t)plane * H * W;

    // ---- zero the padding cells -------------------------------------------
    if (tid < TILE_ROWS) {
        tile[tid * ROW_STRIDE + (PAD_L - 1)] = 0.0f;   // iw = -1
        tile[tid * ROW_STRIDE + (PAD_L + W)] = 0.0f;   // iw = W
    }
    // only the first/last grid rows have an out-of-range input row
    if (ih0 < 0) {
        for (int c = tid; c < W + 2; c += NTHREADS)
            tile[0 * ROW_STRIDE + (PAD_L - 1) + c] = 0.0f;
    }
    if (ih0 + TILE_ROWS - 1 >= H) {
        for (int c = tid; c < W + 2; c += NTHREADS)
            tile[(TILE_ROWS - 1) * ROW_STRIDE + (PAD_L - 1) + c] = 0.0f;
    }

    // ---- async DMA of the in-range tile rows into LDS ---------------------
    // 64 lanes x b128 (4 floats) covers one 256-float row; 256 threads move
    // 4 rows per pass; 5 passes cover 18 rows. Disjoint from the pad cells,
    // so no ordering needed vs the DS stores above.
    const int rsub = tid >> 6;            // 0..3
    const int c4   = (tid & 63) << 2;     // 0,4,...,252
#pragma unroll
    for (int p = 0; p < 5; ++p) {
        int r  = p * 4 + rsub;
        int ih = ih0 + r;
        if (r < TILE_ROWS && ih >= 0 && ih < H) {
            // low 32 bits of a flat shared pointer == LDS byte offset
            unsigned lds_off = (unsigned)(uintptr_t)&tile[r * ROW_STRIDE + PAD_L + c4];
            const float* g = xp + (size_t)ih * W + c4;
            asm volatile("global_load_async_to_lds_b128 %0, %1, off"
                         :: "v"(lds_off), "v"(g) : "memory");
        }
    }
    asm volatile("s_wait_asynccnt 0" ::: "memory");   // my wave's DMAs done
    __syncthreads();                                  // everyone's DMAs + pad stores done

    // ---- 16-tap stencil from LDS ------------------------------------------
    float kk[16];
#pragma unroll
    for (int i = 0; i < 16; ++i) kk[i] = kw[i];       // uniform -> s_load

#pragma unroll
    for (int j = 0; j < 4; ++j) {
        int oi  = tid + j * NTHREADS;                 // 0..1023
        int ohl = oi >> 7;                            // local output row 0..7
        int ow  = oi & 127;                           // output col (coalesced)
        // LDS col for (ow, s): 2*ow - 1 + s + PAD_L
        const float* trow = &tile[(2 * ohl) * ROW_STRIDE + 2 * ow + (PAD_L - 1)];
        float acc = 0.0f;
#pragma unroll
        for (int r = 0; r < 4; ++r)
#pragma unroll
            for (int s = 0; s < 4; ++s)
                acc = fmaf(kk[r * 4 + s], trow[r * ROW_STRIDE + s], acc);
        y[((size_t)plane * OH + (oh0 + ohl)) * OW + ow] = acc;
    }
}

extern "C" void kernel_launch(void* const* d_in, const int* in_sizes, int n_in,
                              void* d_out, int out_size, void* d_ws, size_t ws_size,
                              hipStream_t stream) {
    (void)in_sizes; (void)n_in; (void)d_ws; (void)ws_size; (void)out_size;
    const float* x  = (const float*)d_in[0];   // (8,128,256,256) f32
    const float* kw = (const float*)d_in[1];   // (4,4) f32
    float* y = (float*)d_out;                  // (8,128,128,128) f32

    const int B = 8, C = 128, H = 256, W = 256;
    const int OH = H / 2, OW = W / 2;
    dim3 grid(OH / TILE_OH, B * C);            // (16, 1024)
    blurpool_ds2_kernel<<<grid, NTHREADS, 0, stream>>>(x, kw, y, H, W, OH, OW);
}